// ExiGCNLayer_19782619365928
// MI455X (gfx1250) — compile-verified
//
#include <hip/hip_runtime.h>
#include <hip/hip_bf16.h>

typedef __attribute__((ext_vector_type(2))) float v2f;
typedef __attribute__((ext_vector_type(8))) float v8f;

#define FD 128   // feature dim D == OUT == 128

// ---------------------------------------------------------------------------
// Kernel 1: zero the two accumulator regions (fixed_term slot := Fin acc,
//           B slot := AH acc). new_Z slot is fully overwritten later.
// ---------------------------------------------------------------------------
__global__ __launch_bounds__(256) void zero2_kernel(float* __restrict__ a,
                                                    float* __restrict__ b,
                                                    long n) {
  long i = (long)blockIdx.x * 256 + threadIdx.x;
  if (i < n) { a[i] = 0.0f; b[i] = 0.0f; }
}

// ---------------------------------------------------------------------------
// Kernel 2: adj edges.  AH[dst] += w*X[src];  Fin[dst] += w*dX[src]
// 128 threads per edge: coalesced 512B gather rows, cacheline-local atomics.
// All accumulators live in L2 (51 MB << 192 MB).
// ---------------------------------------------------------------------------
__global__ __launch_bounds__(256) void scatter_adj_kernel(
    const float* __restrict__ X, const float* __restrict__ dX,
    const int* __restrict__ src, const int* __restrict__ dst,
    const float* __restrict__ w,
    float* __restrict__ AH, float* __restrict__ Fin, int E) {
  int e = blockIdx.x * 2 + (threadIdx.x >> 7);
  int j = threadIdx.x & 127;
  if (e < E) {
    long s = src[e];
    long d = dst[e];
    float ww = w[e];
    unsafeAtomicAdd(&AH[d * FD + j], ww * X[s * FD + j]);
    unsafeAtomicAdd(&Fin[d * FD + j], ww * dX[s * FD + j]);
  }
}

// ---------------------------------------------------------------------------
// Kernel 3: dadj edges.  Fin[dst] += w*(X[src] + dX[src])
// ---------------------------------------------------------------------------
__global__ __launch_bounds__(256) void scatter_dadj_kernel(
    const float* __restrict__ X, const float* __restrict__ dX,
    const int* __restrict__ src, const int* __restrict__ dst,
    const float* __restrict__ w,
    float* __restrict__ Fin, int E) {
  int e = blockIdx.x * 2 + (threadIdx.x >> 7);
  int j = threadIdx.x & 127;
  if (e < E) {
    long s = src[e];
    long d = dst[e];
    float ww = w[e];
    unsafeAtomicAdd(&Fin[d * FD + j], ww * (X[s * FD + j] + dX[s * FD + j]));
  }
}

// ---------------------------------------------------------------------------
// Kernel 4: per 16-row tile:
//   B   = AH + Fin                      (written over the AH slot)
//   fx  = Fin @ W                       (V_WMMA_F32_16X16X4_F32, K=128)
//   Z'  = cached_Z + fx + B @ dW + db
// 256 threads = 8 waves; wave w owns output columns [16w, 16w+16) of BOTH
// GEMMs.  LDS row stride 132 floats -> conflict-free A-fragment reads.
// EXEC is all-1s for every WMMA (full block, uniform control flow).
// ---------------------------------------------------------------------------
__global__ __launch_bounds__(256) void gemm_epilogue_kernel(
    const float* __restrict__ cachedZ,
    const float* __restrict__ W,     // [128,128] row-major (K, N)
    const float* __restrict__ dW,    // [128,128]
    const float* __restrict__ dbias, // [128]
    float* __restrict__ outZ,
    float* __restrict__ outFixed,    // holds Fin on entry, fixed_term on exit
    float* __restrict__ outB,        // holds AH on entry, B on exit
    int n) {
  __shared__ float sFin[16 * 132];
  __shared__ float sB[16 * 132];

  const int tid = threadIdx.x;
  const long r0 = (long)blockIdx.x * 16;

  // Stage the 16x128 Fin tile and B = AH + Fin tile into LDS (coalesced).
  for (int i = tid; i < 16 * FD; i += 256) {
    int row = i >> 7, col = i & 127;
    long rr = r0 + row;
    if (rr >= n) rr = n - 1;              // clamp (N divisible by 16 anyway)
    long g = rr * FD + col;
    float fin = outFixed[g];
    float bb = outB[g] + fin;
    sFin[row * 132 + col] = fin;
    sB[row * 132 + col] = bb;
  }
  __syncthreads();

  // Write B out (overwrites the AH accumulator slot, rows owned by this block).
  for (int i = tid; i < 16 * FD; i += 256) {
    int row = i >> 7, col = i & 127;
    if (r0 + row < n) outB[(r0 + row) * FD + col] = sB[row * 132 + col];
  }

  const int wave = tid >> 5;
  const int lane = tid & 31;
  const int half = lane >> 4;      // 0: K pair {0,1}, 1: K pair {2,3} of each step
  const int lr = lane & 15;        // A row (M) for A-frag, N column for B-frag
  const int colBase = wave * 16 + lr;

  v8f accF = {};  // Fin @ W
  v8f accB = {};  // B   @ dW

#pragma unroll
  for (int ks = 0; ks < 32; ++ks) {
    const int kA = ks * 4 + half * 2;
    v2f aF, aB, bW, bD;
    // A fragment (16x4 f32): lane<16 -> K={k0,k0+1}, lane>=16 -> K={k0+2,k0+3}
    aF.x = sFin[lr * 132 + kA];
    aF.y = sFin[lr * 132 + kA + 1];
    aB.x = sB[lr * 132 + kA];
    aB.y = sB[lr * 132 + kA + 1];
    // B fragment (4x16 f32): same K split, N across lanes 0..15 per half
    bW.x = W[(long)kA * FD + colBase];
    bW.y = W[(long)(kA + 1) * FD + colBase];
    bD.x = dW[(long)kA * FD + colBase];
    bD.y = dW[(long)(kA + 1) * FD + colBase];
    accF = __builtin_amdgcn_wmma_f32_16x16x4_f32(
        false, aF, false, bW, (short)0, accF, false, false);
    accB = __builtin_amdgcn_wmma_f32_16x16x4_f32(
        false, aB, false, bD, (short)0, accB, false, false);
  }

  const float bias = dbias[colBase];

  // C/D layout: VGPR v, lane half h -> M = v + 8*h, N = lr.
#pragma unroll
  for (int v = 0; v < 8; ++v) {
    long row = r0 + v + half * 8;
    if (row < n) {
      long idx = row * FD + colBase;
      float fx = accF[v];
      outFixed[idx] = fx;
      outZ[idx] = cachedZ[idx] + fx + accB[v] + bias;
    }
  }
}

// ---------------------------------------------------------------------------
extern "C" void kernel_launch(void* const* d_in, const int* in_sizes, int n_in,
                              void* d_out, int out_size, void* d_ws,
                              size_t ws_size, hipStream_t stream) {
  const float* X   = (const float*)d_in[0];   // features        [N,128]
  const float* dX  = (const float*)d_in[1];   // delta_features  [N,128]
  const int* asrc  = (const int*)d_in[2];     // adj_src  [E]
  const int* adst  = (const int*)d_in[3];     // adj_dst  [E]
  const float* aw  = (const float*)d_in[4];   // adj_w    [E]
  const int* dsrc  = (const int*)d_in[5];     // dadj_src [ED]
  const int* ddst  = (const int*)d_in[6];     // dadj_dst [ED]
  const float* dwt = (const float*)d_in[7];   // dadj_w   [ED]
  const float* cZ  = (const float*)d_in[8];   // cached_Z [N,128]
  const float* W   = (const float*)d_in[9];   // W        [128,128]
  const float* dW  = (const float*)d_in[10];  // delta_W  [128,128]
  const float* db  = (const float*)d_in[11];  // delta_bias [128]

  const int n  = in_sizes[0] / FD;
  const int E  = in_sizes[2];
  const int ED = in_sizes[5];

  float* out      = (float*)d_out;
  float* outZ     = out;                       // new_Z
  float* outFixed = out + (long)n * FD;        // fixed_term (Fin accumulator)
  float* outB     = out + 2L * (long)n * FD;   // B          (AH accumulator)

  const long tot = (long)n * FD;

  zero2_kernel<<<(int)((tot + 255) / 256), 256, 0, stream>>>(outFixed, outB,
                                                             tot);
  scatter_adj_kernel<<<(E + 1) / 2, 256, 0, stream>>>(X, dX, asrc, adst, aw,
                                                      outB, outFixed, E);
  scatter_dadj_kernel<<<(ED + 1) / 2, 256, 0, stream>>>(X, dX, dsrc, ddst, dwt,
                                                        outFixed, ED);
  gemm_epilogue_kernel<<<(n + 15) / 16, 256, 0, stream>>>(cZ, W, dW, db, outZ,
                                                          outFixed, outB, n);
}